// EncoderAttn_53472342835397
// MI455X (gfx1250) — compile-verified
//
#include <hip/hip_runtime.h>
#include <hip/hip_bf16.h>
#include <math.h>

// ---------------------------------------------------------------------------
// EncoderAttn for MI455X (gfx1250, wave32, WMMA).
//
// Shapes: B=16, C=512, N=192, Cs=128, Cr=64, INTER = N*Cr + C = 12800.
//
// Restructure: rel_pre[b,o,i,j] = u[b,i,o] - v[b,j,o] + re_b[o]
//   (u = re_w @ th_act, v = re_w @ ph_act), so the [B,N,N,Cs] diff tensor
// (302 MB) and [B,Cr,N,N] rel tensor (151 MB) are never materialized.
// BN stats of rel_pre come from per-batch sums (incl. cross term S1u*S1v).
// The dominant 40 GFLOP attention GEMM is fused: A-operand rows are
// generated in-register from LDS-resident uu/vv via sub+relu, multiplied
// with L2-resident att_w using V_WMMA_F32_16X16X4_F32 (exact fp32 path).
// ---------------------------------------------------------------------------

#define B_    16
#define C_    512
#define N_    192
#define CS    128
#define CR    64
#define INTER 12800
#define EPSF  1e-5f

typedef float v2f __attribute__((ext_vector_type(2)));
typedef float v8f __attribute__((ext_vector_type(8)));

// ---------------------------------------------------------------------------
// K1: th_pre/ph_pre[b,o,n] = sum_c W[o,c]*x[b,c,n] + bias[o]   (0.4 GFLOP)
// ---------------------------------------------------------------------------
__global__ __launch_bounds__(192) void k_proj(const float* __restrict__ x,
                                              const float* __restrict__ W,
                                              const float* __restrict__ bias,
                                              float* __restrict__ out) {
  __shared__ float wrow[C_];
  const int b = blockIdx.x, o = blockIdx.y;
  const float* wr = W + (size_t)o * C_;
  for (int c = threadIdx.x; c < C_; c += 192) wrow[c] = wr[c];
  __syncthreads();
  const int n = threadIdx.x;
  const float* xb = x + (size_t)b * C_ * N_ + n;
  float acc = 0.f;
  for (int c = 0; c < C_; ++c) acc = fmaf(wrow[c], xb[(size_t)c * N_], acc);
  out[((size_t)b * CS + o) * N_ + n] = acc + bias[o];
}

// ---------------------------------------------------------------------------
// K2: per-channel BN stats over (b,n); emits scale = g*rsqrt(var+eps),
//     shift = be - mean*scale.  (used for theta/phi with CH=128, att CH=512)
// ---------------------------------------------------------------------------
__global__ __launch_bounds__(256) void k_bnstats(const float* __restrict__ pre,
                                                 const float* __restrict__ g,
                                                 const float* __restrict__ be,
                                                 float* __restrict__ scale,
                                                 float* __restrict__ shift,
                                                 int CH) {
  __shared__ float s1[256], s2[256];
  const int ch = blockIdx.x;
  const int M = B_ * N_;
  float a = 0.f, q = 0.f;
  for (int t = threadIdx.x; t < M; t += 256) {
    const int bb = t / N_, n = t - bb * N_;
    const float v = pre[((size_t)bb * CH + ch) * N_ + n];
    a += v;
    q += v * v;
  }
  s1[threadIdx.x] = a;
  s2[threadIdx.x] = q;
  __syncthreads();
  for (int off = 128; off > 0; off >>= 1) {
    if (threadIdx.x < off) {
      s1[threadIdx.x] += s1[threadIdx.x + off];
      s2[threadIdx.x] += s2[threadIdx.x + off];
    }
    __syncthreads();
  }
  if (threadIdx.x == 0) {
    const float mean = s1[0] / (float)M;
    const float var = s2[0] / (float)M - mean * mean;
    const float sc = g[ch] * rsqrtf(var + EPSF);
    scale[ch] = sc;
    shift[ch] = be[ch] - mean * sc;
  }
}

// ---------------------------------------------------------------------------
// K3: u/v[b,n,o] = sum_c re_w[o,c] * relu(pre[b,c,n]*scale[c]+shift[c])
// ---------------------------------------------------------------------------
__global__ __launch_bounds__(64) void k_uv(const float* __restrict__ pre,
                                           const float* __restrict__ scale,
                                           const float* __restrict__ shift,
                                           const float* __restrict__ reW,
                                           float* __restrict__ outUV) {
  __shared__ float xv[CS];
  const int n = blockIdx.x, b = blockIdx.y;
  const int t = threadIdx.x;
  for (int c = t; c < CS; c += 64) {
    const float p = pre[((size_t)b * CS + c) * N_ + n];
    xv[c] = fmaxf(fmaf(p, scale[c], shift[c]), 0.f);
  }
  __syncthreads();
  const float* wr = reW + (size_t)t * CS;
  float acc = 0.f;
  for (int c = 0; c < CS; ++c) acc = fmaf(wr[c], xv[c], acc);
  outUV[((size_t)b * N_ + n) * CR + t] = acc;
}

// ---------------------------------------------------------------------------
// K4: closed-form BN(0,2,3) stats for rel_pre = u - v + re_b:
//   sum_{i,j}(u'_i - v_j)^2 = N*S2u' - 2*S1u'*S1v + N*S2v   (per batch)
// Emits alpha[o] = re_g*rsqrt(var+eps), cu[o] = alpha*(re_b-mean)+re_be
//   so rel_act = relu((alpha*u+cu) - alpha*v).
// ---------------------------------------------------------------------------
__global__ __launch_bounds__(256) void k_relstats(const float* __restrict__ u,
                                                  const float* __restrict__ v,
                                                  const float* __restrict__ reB,
                                                  const float* __restrict__ reG,
                                                  const float* __restrict__ reBe,
                                                  float* __restrict__ alphaOut,
                                                  float* __restrict__ cuOut) {
  __shared__ float s[4][256];
  const int o = blockIdx.x, t = threadIdx.x;
  float A1 = 0.f, A2 = 0.f, Bb1 = 0.f, Bb2 = 0.f, X = 0.f;
  for (int bb = 0; bb < B_; ++bb) {
    float lu = 0.f, lv = 0.f;
    if (t < N_) {
      lu = u[((size_t)bb * N_ + t) * CR + o];
      lv = v[((size_t)bb * N_ + t) * CR + o];
    }
    s[0][t] = lu;
    s[1][t] = lu * lu;
    s[2][t] = lv;
    s[3][t] = lv * lv;
    __syncthreads();
    for (int off = 128; off > 0; off >>= 1) {
      if (t < off)
        for (int q = 0; q < 4; ++q) s[q][t] += s[q][t + off];
      __syncthreads();
    }
    if (t == 0) {
      A1 += s[0][0];
      A2 += s[1][0];
      Bb1 += s[2][0];
      Bb2 += s[3][0];
      X += s[0][0] * s[2][0];  // cross term S1u[b]*S1v[b]
    }
    __syncthreads();
  }
  if (t == 0) {
    const float r = reB[o];
    const float Mbn = (float)(B_ * N_);  // 3072
    const float mean = (A1 - Bb1) / Mbn + r;
    const float sumsq = (float)N_ * A2 + 2.f * r * (float)N_ * A1 +
                        (float)B_ * (float)N_ * (float)N_ * r * r - 2.f * X -
                        2.f * (float)N_ * r * Bb1 + (float)N_ * Bb2;
    const float var = sumsq / (float)(B_ * N_ * N_) - mean * mean;
    const float al = reG[o] * rsqrtf(var + EPSF);
    alphaOut[o] = al;
    cuOut[o] = al * (r - mean) + reBe[o];
  }
}

// ---------------------------------------------------------------------------
// K5: fused attention GEMM (40 GFLOP) using V_WMMA_F32_16X16X4_F32.
//   attPre[b,oc,i] = att_b[oc]
//     + sum_{k<512}  att_w[oc,k] * inp[b,k,i]                (xt part)
//     + sum_{j,o}    att_w[oc,512+j*64+o] * relu(uu[b,i,o]-vv[b,j,o])
// Grid: 16 b x 16 oc-tiles(32). Block 256 (8 waves): wave -> (nt in 0..1,
// mg in 0..3), wave owns 3 M-tiles x 1 N-tile of 16x16 f32 accumulators.
// LDS: uu[192][66] (padded: bank-conflict-free lane-varying-row b64 reads),
//      vv[192][64], bw[32][68] staged att_w column-block for coalescing.
// ---------------------------------------------------------------------------
#define UU_STRIDE 66
#define BW_STRIDE 68
#define ATT_LDS_BYTES ((N_ * UU_STRIDE + N_ * CR + 32 * BW_STRIDE) * 4)

__global__ __launch_bounds__(256) void k_attgemm(
    const float* __restrict__ inp, const float* __restrict__ attW,
    const float* __restrict__ attB, const float* __restrict__ u,
    const float* __restrict__ v, const float* __restrict__ relAlpha,
    const float* __restrict__ relCu, float* __restrict__ attPre) {
  extern __shared__ float lds[];
  float* uu = lds;                   // 192*66
  float* vv = uu + N_ * UU_STRIDE;   // 192*64
  float* bw = vv + N_ * CR;          // 32*68

  const int b = blockIdx.x >> 4;
  const int ocBase = (blockIdx.x & 15) * 32;
  const int tid = threadIdx.x;

  // Stage BN-folded uu/vv for this batch into LDS (one pass, coalesced).
  const float* ub = u + (size_t)b * N_ * CR;
  const float* vb = v + (size_t)b * N_ * CR;
  for (int idx = tid; idx < N_ * CR; idx += 256) {
    const int o = idx & (CR - 1);
    const int i = idx >> 6;
    const float al = relAlpha[o];
    uu[i * UU_STRIDE + o] = fmaf(al, ub[idx], relCu[o]);
    vv[idx] = al * vb[idx];
  }

  const int wave = tid >> 5;
  const int lane = tid & 31;
  const int half = lane >> 4;  // A: K pair select, B: K row select
  const int l16 = lane & 15;
  const int nt = wave & 1;   // N(=oc) tile within block
  const int mg = wave >> 1;  // M tile group (3 tiles each)
  const int oc0 = ocBase + nt * 16;

  const v8f vzero = {0.f, 0.f, 0.f, 0.f, 0.f, 0.f, 0.f, 0.f};
  v8f acc[3] = {vzero, vzero, vzero};

  const float* wbase = attW + (size_t)ocBase * INTER;
  const float* xb = inp + (size_t)b * (C_ * N_);
  const int browoff = (nt * 16 + l16) * BW_STRIDE;

  // 200 column-blocks of 64: cb<8 -> xt part (K=0..511), else rel part.
  for (int cb = 0; cb < 200; ++cb) {
    const int colBase = cb * 64;
    __syncthreads();  // previous bw consumed (also covers uu/vv staging)
    // Coalesced stage of att_w[ocBase..+31, colBase..+63] into LDS.
    for (int idx = tid; idx < 512; idx += 256) {
      const int r = idx >> 4;
      const int c4 = (idx & 15) << 2;
      const float4 t4 =
          *(const float4*)(wbase + (size_t)r * INTER + colBase + c4);
      float* d = bw + r * BW_STRIDE + c4;
      d[0] = t4.x;
      d[1] = t4.y;
      d[2] = t4.z;
      d[3] = t4.w;
    }
    __syncthreads();

    if (cb < 8) {
      // A[i,k] = inp[b,k,i]  (coalesced: consecutive lanes -> consecutive i)
      for (int kk = 0; kk < 64; kk += 4) {
        v2f bf;  // B 4x16: v0 = row(k0+half), v1 = row(k0+2+half)
        bf.x = bw[browoff + kk + half];
        bf.y = bw[browoff + kk + half + 2];
        const int ka = colBase + kk + half * 2;
#pragma unroll
        for (int t = 0; t < 3; ++t) {
          const int i = (mg * 3 + t) * 16 + l16;
          v2f af;  // A 16x4: lane holds K = k0+2*half, k0+2*half+1
          af.x = xb[(size_t)ka * N_ + i];
          af.y = xb[(size_t)(ka + 1) * N_ + i];
          acc[t] = __builtin_amdgcn_wmma_f32_16x16x4_f32(
              false, af, false, bf, (short)0, acc[t], false, false);
        }
      }
    } else {
      const int j = cb - 8;
      const float* vrow = vv + j * CR;
      for (int kk = 0; kk < 64; kk += 4) {
        v2f bf;
        bf.x = bw[browoff + kk + half];
        bf.y = bw[browoff + kk + half + 2];
        const v2f vj = *(const v2f*)(vrow + kk + half * 2);  // broadcast
#pragma unroll
        for (int t = 0; t < 3; ++t) {
          const int i = (mg * 3 + t) * 16 + l16;
          const v2f up = *(const v2f*)(uu + i * UU_STRIDE + kk + half * 2);
          v2f af;  // generated A: relu(uu - vv)
          af.x = fmaxf(up.x - vj.x, 0.f);
          af.y = fmaxf(up.y - vj.y, 0.f);
          acc[t] = __builtin_amdgcn_wmma_f32_16x16x4_f32(
              false, af, false, bf, (short)0, acc[t], false, false);
        }
      }
    }
  }

  // D layout: lane l, comp r -> row i = 8*half + r (contiguous), col oc = l16.
  const int oc = oc0 + l16;
  const float bias = attB[oc];
#pragma unroll
  for (int t = 0; t < 3; ++t) {
    const int ibase = (mg * 3 + t) * 16 + half * 8;
    float* dst = attPre + ((size_t)(b * C_ + oc)) * N_ + ibase;
#pragma unroll
    for (int r = 0; r < 8; ++r) dst[r] = acc[t][r] + bias;
  }
}

// ---------------------------------------------------------------------------
// K6: out[b,c] = sum_n sigmoid(bn(attPre)) * x / sum_n sigmoid(bn(attPre))
// ---------------------------------------------------------------------------
__global__ __launch_bounds__(256) void k_final(const float* __restrict__ attPre,
                                               const float* __restrict__ scale,
                                               const float* __restrict__ shift,
                                               const float* __restrict__ x,
                                               float* __restrict__ out) {
  const int id = blockIdx.x * 256 + threadIdx.x;
  if (id >= B_ * C_) return;
  const int c = id & (C_ - 1);
  const float* ap = attPre + (size_t)id * N_;
  const float* xp = x + (size_t)id * N_;
  const float sc = scale[c], sh = shift[c];
  float num = 0.f, den = 0.f;
  for (int n = 0; n < N_; ++n) {
    const float p = fmaf(ap[n], sc, sh);
    const float a = 1.f / (1.f + expf(-p));
    den += a;
    num = fmaf(a, xp[n], num);
  }
  out[id] = num / den;
}

// ---------------------------------------------------------------------------
// Launch
// ---------------------------------------------------------------------------
extern "C" void kernel_launch(void* const* d_in, const int* in_sizes, int n_in,
                              void* d_out, int out_size, void* d_ws,
                              size_t ws_size, hipStream_t stream) {
  const float* x = (const float*)d_in[0];
  // d_in[1] = retain (0 for this harness -> [B,C] output)
  const float* theta_w = (const float*)d_in[2];
  const float* theta_b = (const float*)d_in[3];
  const float* theta_g = (const float*)d_in[4];
  const float* theta_be = (const float*)d_in[5];
  const float* phi_w = (const float*)d_in[6];
  const float* phi_b = (const float*)d_in[7];
  const float* phi_g = (const float*)d_in[8];
  const float* phi_be = (const float*)d_in[9];
  const float* re_w = (const float*)d_in[10];
  const float* re_b = (const float*)d_in[11];
  const float* re_g = (const float*)d_in[12];
  const float* re_be = (const float*)d_in[13];
  const float* att_w = (const float*)d_in[14];
  const float* att_b = (const float*)d_in[15];
  const float* att_g = (const float*)d_in[16];
  const float* att_be = (const float*)d_in[17];
  float* out = (float*)d_out;

  float* ws = (float*)d_ws;
  float* thPre = ws;                  // 16*128*192 = 393216
  float* phPre = thPre + 393216;      // 393216
  float* thScale = phPre + 393216;    // 128
  float* thShift = thScale + 128;     // 128
  float* phScale = thShift + 128;     // 128
  float* phShift = phScale + 128;     // 128
  float* uBuf = phShift + 128;        // 16*192*64 = 196608
  float* vBuf = uBuf + 196608;        // 196608
  float* relAlpha = vBuf + 196608;    // 64
  float* relCu = relAlpha + 64;       // 64
  float* attPre = relCu + 64;         // 16*512*192 = 1572864
  float* attScale = attPre + 1572864; // 512
  float* attShift = attScale + 512;   // 512  -> total ~10.5 MB

  (void)in_sizes; (void)n_in; (void)out_size; (void)ws_size;

  // Allow >64KB dynamic LDS for the fused GEMM (WGP has 320KB).
  hipFuncSetAttribute((const void*)k_attgemm,
                      hipFuncAttributeMaxDynamicSharedMemorySize,
                      ATT_LDS_BYTES);

  // Stage A: theta/phi projections.
  k_proj<<<dim3(B_, CS), 192, 0, stream>>>(x, theta_w, theta_b, thPre);
  k_proj<<<dim3(B_, CS), 192, 0, stream>>>(x, phi_w, phi_b, phPre);

  // Stage B: BN stats -> scale/shift.
  k_bnstats<<<CS, 256, 0, stream>>>(thPre, theta_g, theta_be, thScale, thShift, CS);
  k_bnstats<<<CS, 256, 0, stream>>>(phPre, phi_g, phi_be, phScale, phShift, CS);

  // Stage C: u = re_w @ relu(bn(th)), v = re_w @ relu(bn(ph)).
  k_uv<<<dim3(N_, B_), 64, 0, stream>>>(thPre, thScale, thShift, re_w, uBuf);
  k_uv<<<dim3(N_, B_), 64, 0, stream>>>(phPre, phScale, phShift, re_w, vBuf);

  // Stage D: closed-form BN stats for rel_pre.
  k_relstats<<<CR, 256, 0, stream>>>(uBuf, vBuf, re_b, re_g, re_be, relAlpha, relCu);

  // Stage E: fused WMMA attention GEMM.
  k_attgemm<<<256, 256, ATT_LDS_BYTES, stream>>>(x, att_w, att_b, uBuf, vBuf,
                                                 relAlpha, relCu, attPre);

  // Stage F: att BN stats + sigmoid-weighted pooled output.
  k_bnstats<<<C_, 256, 0, stream>>>(attPre, att_g, att_be, attScale, attShift, C_);
  k_final<<<(B_ * C_ + 255) / 256, 256, 0, stream>>>(attPre, attScale, attShift, x, out);
}